// GraphNeuralNetwork_19250043420814
// MI455X (gfx1250) — compile-verified
//
#include <hip/hip_runtime.h>

#define DIN   256
#define H_DIM 128
#define DOUT  64
// 1/sqrt(1 + 1e-5) for BatchNorm eval with running_var = 1
#define BN_RSQRT 0.9999950000374996f

typedef __attribute__((ext_vector_type(2))) float v2f;
typedef __attribute__((ext_vector_type(8))) float v8f;

__device__ __forceinline__ v8f wmma4(v2f a, v2f b, v8f c) {
    // D = A(16x4 f32) * B(4x16 f32) + C(16x16 f32); wave32, EXEC all ones
    return __builtin_amdgcn_wmma_f32_16x16x4_f32(
        /*neg_a=*/false, a, /*neg_b=*/false, b,
        /*c_mod=*/(short)0, c, /*reuse_a=*/false, /*reuse_b=*/false);
}

// Accumulate a 32x32 tile: acc[mi][ni] covers rows row0+16*mi.., cols col0+16*ni..
__device__ __forceinline__ void gemm_accum(const float* __restrict__ A,
                                           const float* __restrict__ W,
                                           int K, int ncols,
                                           int row0, int col0, int m, int hi,
                                           float rs0, float rs1,
                                           v8f (&acc)[2][2]) {
    const float* ar0 = A + (size_t)(row0 + m) * K;        // A-frag rows fixed per lane
    const float* ar1 = A + (size_t)(row0 + 16 + m) * K;
    for (int k = 0; k < K; k += 4) {
        const int ka = k + 2 * hi;
        v2f a0, a1, b0, b1;
        a0.x = ar0[ka] * rs0;  a0.y = ar0[ka + 1] * rs0;  // 64-bit contiguous loads
        a1.x = ar1[ka] * rs1;  a1.y = ar1[ka + 1] * rs1;
        const float* wr = W + (size_t)ka * ncols;         // W row-major [K, ncols]
        b0.x = wr[col0 + m];       b0.y = wr[ncols + col0 + m];
        b1.x = wr[col0 + 16 + m];  b1.y = wr[ncols + col0 + 16 + m];
        acc[0][0] = wmma4(a0, b0, acc[0][0]);
        acc[0][1] = wmma4(a0, b1, acc[0][1]);
        acc[1][0] = wmma4(a1, b0, acc[1][0]);
        acc[1][1] = wmma4(a1, b1, acc[1][1]);
    }
}

// out[nrows,ncols] = epilogue( A0[nrows,K0]*rowscale @ W0 + (A1 @ W1) )
// epilogue: v = (v + bias[col]) * (gamma[col]*BN_RSQRT) + beta[col]; optional relu
__global__ __launch_bounds__(128)
void gemm_fused(const float* __restrict__ A0, const float* __restrict__ W0, int K0,
                const float* __restrict__ A1, const float* __restrict__ W1, int K1,
                const float* __restrict__ rowscale,
                const float* __restrict__ bias,
                const float* __restrict__ gamma, const float* __restrict__ beta,
                float* __restrict__ out, int nrows, int ncols, int do_relu) {
    const int wave = threadIdx.x >> 5;
    const int lane = threadIdx.x & 31;
    const int tilesN = ncols >> 5;                 // 32-wide col tiles
    const int tilesM = nrows >> 5;                 // 32-tall row tiles
    const int tile = blockIdx.x * 4 + wave;        // wave-uniform
    if (tile >= tilesM * tilesN) return;
    const int tm = tile / tilesN, tn = tile % tilesN;
    const int row0 = tm * 32, col0 = tn * 32;
    const int m = lane & 15, hi = lane >> 4;

    v8f acc[2][2] = {};

    float rs0 = 1.0f, rs1 = 1.0f;
    if (rowscale) { rs0 = rowscale[row0 + m]; rs1 = rowscale[row0 + 16 + m]; }
    gemm_accum(A0, W0, K0, ncols, row0, col0, m, hi, rs0, rs1, acc);
    if (A1) gemm_accum(A1, W1, K1, ncols, row0, col0, m, hi, 1.0f, 1.0f, acc);

#pragma unroll
    for (int mi = 0; mi < 2; ++mi) {
#pragma unroll
        for (int ni = 0; ni < 2; ++ni) {
            const int col = col0 + ni * 16 + m;
            const float bi = bias ? bias[col] : 0.0f;
            const float sc = gamma ? gamma[col] * BN_RSQRT : 1.0f;
            const float sh = beta ? beta[col] : 0.0f;
#pragma unroll
            for (int r = 0; r < 8; ++r) {
                const int row = row0 + mi * 16 + r + 8 * hi;  // C/D VGPR layout
                float v = acc[mi][ni][r];
                v = (v + bi) * sc + sh;
                if (do_relu) v = fmaxf(v, 0.0f);
                out[(size_t)row * ncols + col] = v;
            }
        }
    }
}

__device__ __forceinline__ void atom_add_f32(float* p, float v) {
    unsafeAtomicAdd(p, v);   // lowers to global_atomic_add_f32 (no return)
}

// One 32-lane group per edge; lane c handles channels [4c, 4c+4)
__global__ __launch_bounds__(256)
void scatter_kernel(const float* __restrict__ h, const int* __restrict__ src,
                    const int* __restrict__ dst, float* __restrict__ agg, int E) {
    const int t = blockIdx.x * blockDim.x + threadIdx.x;
    const int e = t >> 5;
    const int lane = t & 31;
    if (e >= E) return;
    const int s = src[e], d = dst[e];
    const float4 v = *(const float4*)(h + (size_t)s * H_DIM + lane * 4);
    float* ap = agg + (size_t)d * H_DIM + lane * 4;
    atom_add_f32(ap + 0, v.x);
    atom_add_f32(ap + 1, v.y);
    atom_add_f32(ap + 2, v.z);
    atom_add_f32(ap + 3, v.w);
}

__global__ __launch_bounds__(256)
void deg_kernel(const int* __restrict__ dst, float* __restrict__ deg, int E) {
    const int e = blockIdx.x * blockDim.x + threadIdx.x;
    if (e < E) atom_add_f32(&deg[dst[e]], 1.0f);
}

__global__ __launch_bounds__(256)
void invdeg_kernel(float* __restrict__ deg, int N) {
    const int i = blockIdx.x * blockDim.x + threadIdx.x;
    if (i < N) deg[i] = 1.0f / fmaxf(deg[i], 1.0f);
}

extern "C" void kernel_launch(void* const* d_in, const int* in_sizes, int n_in,
                              void* d_out, int out_size, void* d_ws, size_t ws_size,
                              hipStream_t stream) {
    const float* x     = (const float*)d_in[0];
    const int*   ei    = (const int*)d_in[1];
    const float* w_in  = (const float*)d_in[2];
    const float* b_in  = (const float*)d_in[3];
    const float* w_l1  = (const float*)d_in[4];
    const float* b_l1  = (const float*)d_in[5];
    const float* w_r1  = (const float*)d_in[6];
    const float* g1    = (const float*)d_in[7];
    const float* be1   = (const float*)d_in[8];
    const float* w_l2  = (const float*)d_in[9];
    const float* b_l2  = (const float*)d_in[10];
    const float* w_r2  = (const float*)d_in[11];
    const float* g2    = (const float*)d_in[12];
    const float* be2   = (const float*)d_in[13];
    const float* w_out = (const float*)d_in[14];
    const float* b_out = (const float*)d_in[15];

    const int N = in_sizes[0] / DIN;   // 100000 (divisible by 32)
    const int E = in_sizes[1] / 2;     // 1600000
    const int* src = ei;
    const int* dst = ei + E;

    float* ws  = (float*)d_ws;
    float* h0  = ws;                                // [N, H]
    float* agg = ws + (size_t)N * H_DIM;            // [N, H]
    float* h1  = ws + 2 * (size_t)N * H_DIM;        // [N, H]
    float* deg = ws + 3 * (size_t)N * H_DIM;        // [N]  (holds invdeg after prep)

    const int tilesH   = (N / 32) * (H_DIM / 32);   // 12500
    const int tilesOut = (N / 32) * (DOUT / 32);    // 6250

    // Degrees (shared by both layers), then invert in place.
    hipMemsetAsync(deg, 0, (size_t)N * sizeof(float), stream);
    deg_kernel<<<(E + 255) / 256, 256, 0, stream>>>(dst, deg, E);
    invdeg_kernel<<<(N + 255) / 256, 256, 0, stream>>>(deg, N);

    // h0 = relu(x @ w_in + b_in)
    gemm_fused<<<(tilesH + 3) / 4, 128, 0, stream>>>(
        x, w_in, DIN, nullptr, nullptr, 0, nullptr, b_in, nullptr, nullptr,
        h0, N, H_DIM, 1);

    // Layer 1: mean-aggregate h0, then h1 = relu(bn(mean@w_l1 + b_l1 + h0@w_r1))
    hipMemsetAsync(agg, 0, (size_t)N * H_DIM * sizeof(float), stream);
    scatter_kernel<<<(E * 32 + 255) / 256, 256, 0, stream>>>(h0, src, dst, agg, E);
    gemm_fused<<<(tilesH + 3) / 4, 128, 0, stream>>>(
        agg, w_l1, H_DIM, h0, w_r1, H_DIM, deg, b_l1, g1, be1,
        h1, N, H_DIM, 1);

    // Layer 2: mean-aggregate h1, then h2 (reuse h0 buffer)
    hipMemsetAsync(agg, 0, (size_t)N * H_DIM * sizeof(float), stream);
    scatter_kernel<<<(E * 32 + 255) / 256, 256, 0, stream>>>(h1, src, dst, agg, E);
    gemm_fused<<<(tilesH + 3) / 4, 128, 0, stream>>>(
        agg, w_l2, H_DIM, h1, w_r2, H_DIM, deg, b_l2, g2, be2,
        h0, N, H_DIM, 1);

    // out = h2 @ w_out + b_out
    gemm_fused<<<(tilesOut + 3) / 4, 128, 0, stream>>>(
        h0, w_out, H_DIM, nullptr, nullptr, 0, nullptr, b_out, nullptr, nullptr,
        (float*)d_out, N, DOUT, 0);
}